// ObjectLoss_14370960573188
// MI455X (gfx1250) — compile-verified
//
#include <hip/hip_runtime.h>
#include <math.h>

// Problem constants (from reference): B=32, A=3, H=64, W=64, C=85, T=50
#define B_  32
#define A_  3
#define H_  64
#define W_  64
#define C_  85
#define T_  50
#define N_GT (B_ * A_ * H_ * W_)   // 393216
#define NBLK 512                    // 512*256 = 131072 threads; N_GT = 3 * 131072 (uniform trip count)
#define NTHR 256                    // 8 wave32 per block
#define THRESH 0.5f

typedef float v2f __attribute__((ext_vector_type(2)));
typedef float v8f __attribute__((ext_vector_type(8)));

// ---------------------------------------------------------------------------
// Wave32 sum-reduction on the WMMA unit (V_WMMA_F32_16X16X4_F32).
// Inputs: two per-lane partials (p0, p1). Output: sum over all 32 lanes of
// (p0 + p1), broadcast to every lane.
//
// Step 1: A(16x4) holds the partials (VGPR0=p0, VGPR1=p1; lanes 16-31 supply
//         K=2,3 per the ISA A-layout), B = ones(4x16):
//         D1[m][n] = p0[m] + p1[m] + p0[m+16] + p1[m+16]  =: r[m]
// Step 2: A = ones(16x4); B(4x16) built from D1 VGPR pairs. D1 VGPR v holds
//         r[v] in lanes 0-15 and r[v+8] in lanes 16-31, which is exactly the
//         B-layout rows {K=0,K=2} / {K=1,K=3}. Four accumulating WMMAs fold
//         all 16 r[m] into C: C[m][n] = sum_m r[m] = total.
// ---------------------------------------------------------------------------
__device__ __forceinline__ float wave_reduce_wmma(float p0, float p1) {
    v2f ones2; ones2.x = 1.0f; ones2.y = 1.0f;
    v2f a;     a.x = p0;       a.y = p1;
    v8f zero8 = {};

    v8f d1 = __builtin_amdgcn_wmma_f32_16x16x4_f32(
        false, a, false, ones2, (short)0, zero8, false, false);

    v8f acc = {};
    v2f b;
    b.x = d1[0]; b.y = d1[1];
    acc = __builtin_amdgcn_wmma_f32_16x16x4_f32(false, ones2, false, b, (short)0, acc, false, false);
    b.x = d1[2]; b.y = d1[3];
    acc = __builtin_amdgcn_wmma_f32_16x16x4_f32(false, ones2, false, b, (short)0, acc, false, false);
    b.x = d1[4]; b.y = d1[5];
    acc = __builtin_amdgcn_wmma_f32_16x16x4_f32(false, ones2, false, b, (short)0, acc, false, false);
    b.x = d1[6]; b.y = d1[7];
    acc = __builtin_amdgcn_wmma_f32_16x16x4_f32(false, ones2, false, b, (short)0, acc, false, false);

    return acc[0];   // total, identical in every lane
}

// ---------------------------------------------------------------------------
// Kernel 1: zero the gt buffer (must happen every call; ws is not re-poisoned)
// ---------------------------------------------------------------------------
__global__ void objloss_zero_gt(float* __restrict__ gt) {
    int tid = blockIdx.x * blockDim.x + threadIdx.x;
    int stride = gridDim.x * blockDim.x;
    for (int i = tid; i < N_GT; i += stride) gt[i] = 0.0f;
}

// ---------------------------------------------------------------------------
// Kernel 2: per-target anchor argmax + scatter of 1.0 (idempotent, no atomics
// needed: every writer to a cell writes the same value after gt was zeroed).
// ---------------------------------------------------------------------------
__global__ void objloss_scatter(const float* __restrict__ targets,
                                const float* __restrict__ anchors,
                                float* __restrict__ gt) {
    int idx = blockIdx.x * blockDim.x + threadIdx.x;
    if (idx >= B_ * T_) return;
    int b = idx / T_;
    const float* t = targets + (size_t)idx * 5;   // [cls, x, y, w, h]
    float tx = t[1], ty = t[2];
    float tw = t[3] * (float)W_;
    float th = t[4] * (float)H_;

    int ti = (int)(tx * (float)W_);  ti = min(max(ti, 0), W_ - 1);
    int tj = (int)(ty * (float)H_);  tj = min(max(tj, 0), H_ - 1);

    float area_t = tw * th;
    float best = -1.0f;
    int   ta   = 0;
    #pragma unroll
    for (int a = 0; a < A_; ++a) {
        float aw = anchors[a * 2 + 0];
        float ah = anchors[a * 2 + 1];
        float inter = fminf(aw, tw) * fminf(ah, th);
        float uni   = aw * ah + area_t - inter;
        float iou   = inter / uni;
        if (iou > best) { best = iou; ta = a; }   // first-max tie-break, like jnp.argmax
    }
    if (best > THRESH) {
        gt[((b * A_ + ta) * H_ + tj) * W_ + ti] = 1.0f;
    }
}

// ---------------------------------------------------------------------------
// Kernel 3: BCE partial sums. Strided gather of output[...,4], per-lane
// accumulation into two registers, WMMA wave reduction, LDS across 8 waves,
// one partial per block (deterministic).
// ---------------------------------------------------------------------------
__global__ void objloss_partial(const float* __restrict__ output,
                                const float* __restrict__ gt,
                                float* __restrict__ partial) {
    int tid    = blockIdx.x * blockDim.x + threadIdx.x;
    int stride = gridDim.x * blockDim.x;

    float s0 = 0.0f, s1 = 0.0f;
    int par = 0;
    for (int i = tid; i < N_GT; i += stride) {     // uniform: N_GT = 3*stride
        float p = output[(size_t)i * C_ + 4];       // strided gather, channel 4
        float g = gt[i];                            // exactly 0.0 or 1.0
        float term = (g > 0.5f) ? logf(p) : log1pf(-p);
        if (par) s1 += term; else s0 += term;
        par ^= 1;
    }

    // EXEC is all-ones here (no divergent exits) -> WMMA is legal.
    float waveSum = wave_reduce_wmma(s0, s1);

    __shared__ float wsum[NTHR / 32];
    int lane = threadIdx.x & 31;
    int wv   = threadIdx.x >> 5;
    if (lane == 0) wsum[wv] = waveSum;
    __syncthreads();
    if (threadIdx.x == 0) {
        float t = 0.0f;
        #pragma unroll
        for (int k = 0; k < NTHR / 32; ++k) t += wsum[k];
        partial[blockIdx.x] = t;
    }
}

// ---------------------------------------------------------------------------
// Kernel 4: deterministic serial finish in double, write -mean.
// ---------------------------------------------------------------------------
__global__ void objloss_finalize(const float* __restrict__ partial,
                                 float* __restrict__ out) {
    double s = 0.0;
    for (int i = 0; i < NBLK; ++i) s += (double)partial[i];
    out[0] = (float)(-s / (double)N_GT);
}

extern "C" void kernel_launch(void* const* d_in, const int* in_sizes, int n_in,
                              void* d_out, int out_size, void* d_ws, size_t ws_size,
                              hipStream_t stream) {
    (void)in_sizes; (void)n_in; (void)out_size; (void)ws_size;
    const float* output  = (const float*)d_in[0];   // (32,3,64,64,85) f32
    const float* targets = (const float*)d_in[1];   // (32,50,5)       f32
    const float* anchors = (const float*)d_in[2];   // (3,2)           f32

    float* gt      = (float*)d_ws;                  // N_GT floats (1.57 MB)
    float* partial = gt + N_GT;                     // NBLK floats
    float* out     = (float*)d_out;                 // scalar loss

    objloss_zero_gt<<<NBLK, NTHR, 0, stream>>>(gt);
    objloss_scatter<<<(B_ * T_ + NTHR - 1) / NTHR, NTHR, 0, stream>>>(targets, anchors, gt);
    objloss_partial<<<NBLK, NTHR, 0, stream>>>(output, gt, partial);
    objloss_finalize<<<1, 1, 0, stream>>>(partial, out);
}